// MyModel_61933428413984
// MI455X (gfx1250) — compile-verified
//
#include <hip/hip_runtime.h>

// out[b,h,q,kv] = scores[b,h,q,kv] + bias[offset[q]]
// B=2, H=16, W=2048  ->  rows = B*H*W = 65536, each row = 2048 fp32 = 8 KiB.
//
// Pure streaming bandwidth kernel: 512 MiB in + 512 MiB out per call.
// One block per row; 256 threads x 2 float4 = 2048 floats = one row.
// Non-temporal hints on both streams (working set >> 192 MB L2).

typedef float f32x4 __attribute__((ext_vector_type(4)));

namespace {
constexpr int kW      = 2048;             // row length (kv dim) and seq len
constexpr int kThreads = 256;             // 8 waves (wave32)
constexpr int kVecPerRow = kW / 4;        // 512 float4 per row
constexpr int kVecPerThread = kVecPerRow / kThreads;  // 2
}

__global__ __launch_bounds__(kThreads)
void bias_row_add_kernel(const float* __restrict__ scores,
                         const float* __restrict__ bias,
                         const int*   __restrict__ offset,
                         float*       __restrict__ out)
{
    const unsigned row = blockIdx.x;          // 0 .. B*H*W-1
    const unsigned q   = row & (kW - 1);      // wave-uniform -> scalar loads

    // Two dependent uniform loads per block (per 8 KiB of traffic): negligible.
    const int   o  = offset[q];
    const float bq = bias[o];

    const size_t base = (size_t)row * (size_t)kW;
    const f32x4* __restrict__ src = (const f32x4*)(scores + base);
    f32x4*       __restrict__ dst = (f32x4*)(out + base);

    const unsigned t = threadIdx.x;

    // Coalesced 128-bit NT load -> add -> NT store, unrolled x2.
    f32x4 v0 = __builtin_nontemporal_load(&src[t]);
    f32x4 v1 = __builtin_nontemporal_load(&src[t + kThreads]);

    v0 += bq;
    v1 += bq;

    __builtin_nontemporal_store(v0, &dst[t]);
    __builtin_nontemporal_store(v1, &dst[t + kThreads]);
}

extern "C" void kernel_launch(void* const* d_in, const int* in_sizes, int n_in,
                              void* d_out, int out_size, void* d_ws, size_t ws_size,
                              hipStream_t stream) {
    (void)in_sizes; (void)n_in; (void)d_ws; (void)ws_size; (void)out_size;

    const float* scores = (const float*)d_in[0];
    const float* bias   = (const float*)d_in[1];
    const int*   offset = (const int*)d_in[2];
    float*       out    = (float*)d_out;

    // rows = B*H*W = 2*16*2048 = 65536
    const unsigned rows = 2u * 16u * 2048u;
    bias_row_add_kernel<<<dim3(rows), dim3(kThreads), 0, stream>>>(scores, bias, offset, out);
}